// QuantumEstimator_65481071402760
// MI455X (gfx1250) — compile-verified
//
#include <hip/hip_runtime.h>

// out[i][j] = cos(x[i][0]) for x: [N, 16] f32, out: [N, 16] f32.
// Pure streaming kernel: ~512 MB total HBM traffic -> ~22 us @ 23.3 TB/s.
// CDNA5 specifics used: NT temporal hints (stream-once vs 192MB L2),
// wave32 shuffle redistribution for fully-coalesced b128 NT stores,
// gfx1250 global_prefetch_b8.

typedef float v4f __attribute__((ext_vector_type(4)));

__global__ __launch_bounds__(256) void cos_bcast_kernel(
    const float* __restrict__ x, float* __restrict__ out, int n_rows) {
  const int lane = (int)(threadIdx.x & 31u);
  const long wave = (long)((blockIdx.x * blockDim.x + threadIdx.x) >> 5);
  const long base_row = wave << 5;              // 32 rows per wave
  if (base_row >= (long)n_rows) return;         // wave-uniform exit: EXEC stays full below

  // ---- load phase: lane l owns row base_row + l ----
  long row = base_row + lane;
  long lrow = row < (long)n_rows ? row : (long)n_rows - 1;  // clamp (no-op for exact grids)

  // NT load of column 0 (row stride 64B). Whole cachelines stream through
  // regardless; NT keeps them from displacing anything in the 192MB L2.
  float v = __builtin_nontemporal_load(x + lrow * 16);

  // gfx1250 prefetch path: pull the stream ~256KB ahead (speculative; OOB dropped).
  __builtin_prefetch(x + (lrow + 4096) * 16, 0, 0);

  // One accurate cos per row (ocml path); compute is ~10x under the BW floor.
  float c = cosf(v);

  // ---- store phase: 4 rounds; each round 8 rows x 64B = one contiguous 512B
  // span per wave (4 full cachelines per global_store_b128 round). Value is
  // fetched from the owning lane via wave32 shuffle so EXEC-full b128 NT
  // stores are perfectly coalesced.
  const int quad = lane & 3;            // which 16B chunk of the 64B row
  const int sub  = lane >> 2;           // which of 8 rows this round
#pragma unroll
  for (int r = 0; r < 4; ++r) {
    const int src = r * 8 + sub;                    // lane owning the row
    const float s = __shfl(c, src, 32);             // ds_bpermute (wave32)
    const long orow = base_row + src;
    if (orow < (long)n_rows) {
      v4f pkt = {s, s, s, s};
      __builtin_nontemporal_store(pkt, (v4f*)(out + orow * 16 + quad * 4));
    }
  }
}

extern "C" void kernel_launch(void* const* d_in, const int* in_sizes, int n_in,
                              void* d_out, int out_size, void* d_ws, size_t ws_size,
                              hipStream_t stream) {
  const float* x = (const float*)d_in[0];
  float* out = (float*)d_out;

  const int n_rows = out_size / 16;               // [N, 16] output
  const int threads = 256;                        // 8 wave32 waves / block
  const int rows_per_block = 256;                 // 32 rows per wave * 8 waves
  const int blocks = (n_rows + rows_per_block - 1) / rows_per_block;

  cos_bcast_kernel<<<blocks, threads, 0, stream>>>(x, out, n_rows);
}